// FLoss_v2_60112362275593
// MI455X (gfx1250) — compile-verified
//
#include <hip/hip_runtime.h>
#include <hip/hip_bf16.h>

#define CC   80
#define DD   8525
#define BB   64
#define EPSF 1e-6f

typedef float v2f __attribute__((ext_vector_type(2)));
typedef float v8f __attribute__((ext_vector_type(8)));

// ---------------------------------------------------------------- workspace
__global__ void zero_ws_kernel(float* ws, int n) {
    int i = blockIdx.x * blockDim.x + threadIdx.x;
    if (i < n) ws[i] = 0.0f;
}

// ---------------------------------------------------------------- cls focal
// Per cls element (d,c): p = sigmoid(x); if g==1: -0.25*(1-p)^2*log(p+eps)
//                                       else:    -0.75*p^2*log(1-p+eps)
// (exact algebraic fusion of l_pos+l_neg since masks are complementary 0/1)
__device__ __forceinline__ float cls_term(const float* __restrict__ ob,
                                          const float* __restrict__ gb,
                                          int i, int E) {
    int ii = (i < E) ? i : 0;                 // clamp so loads stay in range,
    int d  = (int)((unsigned)ii / (unsigned)CC); // keep EXEC all-ones for WMMA
    float x = ob[ii + d * 5];                 // d*85 + c
    float g = gb[ii + d * 8];                 // d*88 + c
    float p = __fdividef(1.0f, 1.0f + __expf(-x));
    bool  pos  = (g == 1.0f);
    float q    = pos ? (p + EPSF) : (1.0f - p + EPSF);
    float w    = pos ? (1.0f - p) : p;
    float coef = (pos ? -0.25f : -0.75f) * w * w;
    float t    = coef * __logf(q);
    return (i < E) ? t : 0.0f;
}

__global__ __launch_bounds__(256)
void cls_kernel(const float* __restrict__ outs, const float* __restrict__ gres,
                float* __restrict__ ws) {
    const int E = DD * CC;                    // 682000 cls elements per batch
    const int b = blockIdx.y;
    const float* ob = outs + (size_t)b * DD * (CC + 5);
    const float* gb = gres + (size_t)b * DD * (CC + 8);

    const int T = blockDim.x * gridDim.x;     // threads per batch slice
    int i = blockIdx.x * blockDim.x + threadIdx.x;
    const int nIter = (E + 2 * T - 1) / (2 * T);  // uniform across block

    v8f acc = {0.f, 0.f, 0.f, 0.f, 0.f, 0.f, 0.f, 0.f};
    v2f ones = {1.0f, 1.0f};                  // B matrix = ones -> row sums

    for (int k = 0; k < nIter; ++k) {
        v2f a;
        a.x = cls_term(ob, gb, i, E);
        a.y = cls_term(ob, gb, i + T, E);
        // D[m][n] = sum_k A[m][k] * 1 + C[m][n]  : fp32 accumulation in XDL
        acc = __builtin_amdgcn_wmma_f32_16x16x4_f32(
            /*neg_a=*/false, a, /*neg_b=*/false, ones,
            /*c_mod=*/(short)0, acc, /*reuse_a=*/false, /*reuse_b=*/false);
        i += 2 * T;
    }

    // D columns are replicated 16x; sum of all lanes' column sums = 16*total.
    float s = acc[0] + acc[1] + acc[2] + acc[3] + acc[4] + acc[5] + acc[6] + acc[7];

    __shared__ float red[256];
    red[threadIdx.x] = s;
    __syncthreads();
    for (int off = 128; off > 0; off >>= 1) {
        if (threadIdx.x < off) red[threadIdx.x] += red[threadIdx.x + off];
        __syncthreads();
    }
    if (threadIdx.x == 0) atomicAdd(&ws[b], red[0] * 0.0625f);
}

// ---------------------------------------------------------------- conf + reg
__global__ __launch_bounds__(256)
void confreg_kernel(const float* __restrict__ outs, const float* __restrict__ gres,
                    const float* __restrict__ grids, float* __restrict__ ws) {
    const int b = blockIdx.y;
    const int d = blockIdx.x * blockDim.x + threadIdx.x;

    float confs = 0.0f, regs = 0.0f, npos = 0.0f;
    if (d < DD) {
        const float* o = outs + ((size_t)b * DD + d) * (CC + 5) + CC;
        const float* g = gres + ((size_t)b * DD + d) * (CC + 8) + CC;

        // confidence BCE on masked anchors
        float pconf = __fdividef(1.0f, 1.0f + __expf(-o[0]));
        float mpp   = (g[5] == 1.0f) ? 1.0f : 0.0f;
        confs = -__logf(pconf + EPSF) * mpp;

        // GIoU regression on positive anchors
        float gx = grids[2 * d], gy = grids[2 * d + 1];
        float p0 = gx - o[1], p1 = gy - o[2], p2 = gx + o[3], p3 = gy + o[4];
        float g0 = g[1], g1 = g[2], g2 = g[3], g3 = g[4];
        float ix1 = fmaxf(p0, g0), iy1 = fmaxf(p1, g1);
        float ix2 = fminf(p2, g2), iy2 = fminf(p3, g3);
        float inter = fmaxf(ix2 - ix1, 0.0f) * fmaxf(iy2 - iy1, 0.0f);
        float ap = fmaxf(p2 - p0, 0.0f) * fmaxf(p3 - p1, 0.0f);
        float ag = fmaxf(g2 - g0, 0.0f) * fmaxf(g3 - g1, 0.0f);
        float un = ap + ag - inter;
        float iou = __fdividef(inter, un + EPSF);
        float cx1 = fminf(p0, g0), cy1 = fminf(p1, g1);
        float cx2 = fmaxf(p2, g2), cy2 = fmaxf(p3, g3);
        float ca = (cx2 - cx1) * (cy2 - cy1);
        float giou = iou - __fdividef(ca - un, ca + EPSF);
        float mpos = (g[6] == 1.0f) ? 1.0f : 0.0f;
        regs = (1.0f - giou) * g[0] * mpos;
        npos = mpos;
    }

    __shared__ float r0[256], r1[256], r2[256];
    r0[threadIdx.x] = confs; r1[threadIdx.x] = regs; r2[threadIdx.x] = npos;
    __syncthreads();
    for (int off = 128; off > 0; off >>= 1) {
        if (threadIdx.x < off) {
            r0[threadIdx.x] += r0[threadIdx.x + off];
            r1[threadIdx.x] += r1[threadIdx.x + off];
            r2[threadIdx.x] += r2[threadIdx.x + off];
        }
        __syncthreads();
    }
    if (threadIdx.x == 0) {
        atomicAdd(&ws[b],      5.0f * r0[0]);   // conf already scaled by 5
        atomicAdd(&ws[BB],     r1[0]);          // global reg sum
        atomicAdd(&ws[BB + 1], r2[0]);          // global n_pos
    }
}

// ---------------------------------------------------------------- finalize
__global__ void finalize_kernel(const float* __restrict__ ws,
                                const int* __restrict__ nums_pos,
                                float* __restrict__ out) {
    __shared__ float red[BB];
    int t = threadIdx.x;
    red[t] = ws[t] / (float)nums_pos[t];        // (cls + 5*conf) / np_f
    __syncthreads();
    for (int off = BB / 2; off > 0; off >>= 1) {
        if (t < off) red[t] += red[t + off];
        __syncthreads();
    }
    if (t == 0) {
        float npos = fmaxf(ws[BB + 1], 1.0f);
        out[0] = red[0] * (1.0f / (float)BB) + 5.0f * ws[BB] / npos;
    }
}

// ---------------------------------------------------------------- launcher
extern "C" void kernel_launch(void* const* d_in, const int* in_sizes, int n_in,
                              void* d_out, int out_size, void* d_ws, size_t ws_size,
                              hipStream_t stream) {
    const float* outs     = (const float*)d_in[0];
    const float* gres     = (const float*)d_in[1];
    const float* grids    = (const float*)d_in[2];
    const int*   nums_pos = (const int*)d_in[3];
    float* out = (float*)d_out;
    float* ws  = (float*)d_ws;   // [0..B-1] per-batch sums, [B] reg, [B+1] n_pos

    zero_ws_kernel<<<1, 128, 0, stream>>>(ws, BB + 2);

    dim3 clsGrid(64, BB);                 // 4096 blocks, 21 WMMA iters/thread
    cls_kernel<<<clsGrid, 256, 0, stream>>>(outs, gres, ws);

    dim3 crGrid((DD + 255) / 256, BB);
    confreg_kernel<<<crGrid, 256, 0, stream>>>(outs, gres, grids, ws);

    finalize_kernel<<<1, BB, 0, stream>>>(ws, nums_pos, out);
}